// TADBoundaryReward_89781996356203
// MI455X (gfx1250) — compile-verified
//
#include <hip/hip_runtime.h>
#include <hip/hip_bf16.h>
#include <math.h>

#define Hh 512
#define Ww 512
#define Bb 4
#define Cc 32
#define NPIX (512*512)
#define ES_K 26214          // int(0.1*512*512)
#define CP_K 51             // 512//10
#define WIN 5
#define EPSF 1e-8f
#define TIE_CAP 2048

typedef float v2f __attribute__((ext_vector_type(2)));
typedef float v8f __attribute__((ext_vector_type(8)));

struct SelState { unsigned prefix; int krem; unsigned thr; int ties; };

// ---------------- exact f32 wave reduction via V_WMMA_F32_16X16X4_F32 ----------------
// A (16x4 f32) holds 2 values/lane; B = all-ones => D[m][n] = row-sum.  Summing the 8 D
// VGPRs gives sum of rows 0-7 on lanes 0-15 and rows 8-15 on lanes 16-31; one xor-16
// exchange yields the exact 64-value wave sum in every lane.
__device__ __forceinline__ float wave_sum2(float a0, float a1) {
#if defined(__gfx1250__) && __has_builtin(__builtin_amdgcn_wmma_f32_16x16x4_f32)
  v2f A; A[0] = a0; A[1] = a1;
  v2f Bm; Bm[0] = 1.0f; Bm[1] = 1.0f;
  v8f Cm = {};
  v8f D = __builtin_amdgcn_wmma_f32_16x16x4_f32(false, A, false, Bm, (short)0, Cm, false, false);
  float s = ((D[0] + D[1]) + (D[2] + D[3])) + ((D[4] + D[5]) + (D[6] + D[7]));
  s += __shfl_xor(s, 16, 32);
  return s;
#else
  float s = a0 + a1;
  #pragma unroll
  for (int off = 16; off > 0; off >>= 1) s += __shfl_xor(s, off, 32);
  return s;
#endif
}

// Block-wide sum (valid on thread 0). Must be called at a block-uniform point.
__device__ __forceinline__ float block_sum2(float a0, float a1, float* lds8) {
  float s = wave_sum2(a0, a1);
  int wid = threadIdx.x >> 5, lane = threadIdx.x & 31;
  if (lane == 0) lds8[wid] = s;
  __syncthreads();
  float r = 0.f;
  if (threadIdx.x == 0) {
    int nw = blockDim.x >> 5;
    for (int i = 0; i < nw; ++i) r += lds8[i];
  }
  __syncthreads();
  return r;
}

// ---------------- value generators (must be bit-identical across kernels) ----------------
__device__ __forceinline__ float seg_at(const float* sb, int h, int w) {
  if ((unsigned)h >= 512u || (unsigned)w >= 512u) return 0.f;
  return sb[h * 512 + w];
}
__device__ __forceinline__ float gm_val(const float* sb, int h, int w) {
  float a = seg_at(sb, h - 1, w - 1), b = seg_at(sb, h - 1, w), c = seg_at(sb, h - 1, w + 1);
  float d = seg_at(sb, h, w - 1),                                e = seg_at(sb, h, w + 1);
  float f = seg_at(sb, h + 1, w - 1), g = seg_at(sb, h + 1, w), i = seg_at(sb, h + 1, w + 1);
  float gx = (c + 2.f * e + i) - (a + 2.f * d + f);
  float gy = (f + 2.f * g + i) - (a + 2.f * b + c);
  // force one fixed contraction shape so every kernel reproduces identical bits
  return sqrtf(fmaf(gx, gx, fmaf(gy, gy, 1e-8f)));
}
__device__ __forceinline__ float edge_val(const float* sb, int h, int w) {
  float s = sb[h * 512 + w];
  float eh = (h < 511) ? fabsf(sb[(h + 1) * 512 + w] - s) : 0.f;
  float ev = (w < 511) ? fabsf(sb[h * 512 + w + 1] - s) : 0.f;
  return fmaxf(eh, ev);
}
template <int SRC>
__device__ __forceinline__ float sel_val(const float* sb, int idx) {
  int h = idx >> 9, w = idx & 511;
  return (SRC == 0) ? gm_val(sb, h, w) : edge_val(sb, h, w);
}

// ---------------- init ----------------
__global__ void init_kernel(double* acc, SelState* es_st, SelState* cp_st,
                            unsigned* hist, int* selCnt, int* tieCnt, int* selFinal) {
  int t = threadIdx.x;
  if (t < 24) acc[t] = 0.0;
  if (t < Bb) {
    es_st[t] = {0u, ES_K, 0u, 0};
    cp_st[t] = {0u, CP_K, 0u, 0};
    selCnt[t] = 0; tieCnt[t] = 0; selFinal[t] = 0;
  }
  for (int i = t; i < Bb * 2048; i += blockDim.x) hist[i] = 0;
}

// ---------------- part 1: internal consistency ----------------
__global__ void ic_kernel(const float* __restrict__ seg, const float* __restrict__ hic,
                          double* ic_sum, double* ic_sumsq) {
  __shared__ float row[512];
  __shared__ float lds8[8];
  int bh = blockIdx.x;
  int b = bh >> 9;
  const float* hr = hic + (size_t)bh * 512;
  const float* sr = seg + (size_t)bh * 512;
  int t = threadIdx.x;
  row[t] = hr[t]; row[t + 256] = hr[t + 256];
  __syncthreads();
  float x0 = 0.f, x1 = 0.f;
  #pragma unroll
  for (int rep = 0; rep < 2; ++rep) {
    int w = t + rep * 256;
    float down = 0.f, up = 0.f;
    #pragma unroll
    for (int i = 1; i <= WIN; ++i) { down += row[(w - i) & 511]; up += row[(w + i) & 511]; }
    float di = (down - up) / (up + down + EPSF);
    float tad = (sr[w] > 0.5f) ? 1.0f : 0.0f;
    float x = di * tad;
    if (rep == 0) x0 = x; else x1 = x;
  }
  float s  = block_sum2(x0, x1, lds8);
  float s2 = block_sum2(x0 * x0, x1 * x1, lds8);
  if (t == 0) { atomicAdd(&ic_sum[b], (double)s); atomicAdd(&ic_sumsq[b], (double)s2); }
}

// ---------------- radix-select (positive floats: bit order == value order) ----------------
template <int SRC, int PASS>
__global__ void hist_kernel(const float* __restrict__ seg, unsigned* __restrict__ hist,
                            const SelState* __restrict__ st) {
  __shared__ unsigned lh[2048];
  int b = blockIdx.y;
  const float* sb = seg + (size_t)b * NPIX;
  for (int i = threadIdx.x; i < 2048; i += blockDim.x) lh[i] = 0;
  __syncthreads();
  unsigned pfx = st[b].prefix;
  int stride = gridDim.x * blockDim.x;
  for (int idx = blockIdx.x * blockDim.x + threadIdx.x; idx < NPIX; idx += stride) {
    unsigned bits = __float_as_uint(sel_val<SRC>(sb, idx));
    if (PASS == 0) atomicAdd(&lh[bits >> 21], 1u);
    else if (PASS == 1) { if ((bits >> 21) == pfx) atomicAdd(&lh[(bits >> 10) & 2047u], 1u); }
    else               { if ((bits >> 10) == pfx) atomicAdd(&lh[bits & 1023u], 1u); }
  }
  __syncthreads();
  for (int i = threadIdx.x; i < 2048; i += blockDim.x) {
    unsigned c = lh[i];
    if (c) atomicAdd(&hist[b * 2048 + i], c);
  }
}

template <int PASS>
__global__ void scan_kernel(unsigned* hist, SelState* st) {
  int t = threadIdx.x;
  if (t < Bb) {
    const int nb = (PASS == 2) ? 1024 : 2048;
    int krem = st[t].krem;
    unsigned* hb = hist + t * 2048;
    int cum = 0, selbin = 0, before = 0;
    for (int bin = nb - 1; bin >= 0; --bin) {
      int c = (int)hb[bin];
      if (cum + c >= krem) { selbin = bin; before = cum; break; }
      cum += c;
    }
    int nk = krem - before;   // elements still needed inside the selected bin
    if (PASS == 0)      { st[t].prefix = (unsigned)selbin;                          st[t].krem = nk; }
    else if (PASS == 1) { st[t].prefix = (st[t].prefix << 11) | (unsigned)selbin;   st[t].krem = nk; }
    else                { st[t].thr    = (st[t].prefix << 10) | (unsigned)selbin;   st[t].ties = nk; }
  }
  __syncthreads();
  for (int i = t; i < Bb * 2048; i += blockDim.x) hist[i] = 0;   // ready for next pass
}

// ---------------- part 2: edge significance masked mean ----------------
__global__ void es_final(const float* __restrict__ seg, const SelState* __restrict__ st,
                         double* es_sum, double* es_cnt) {
  __shared__ float lds8[8];
  int b = blockIdx.y;
  const float* sb = seg + (size_t)b * NPIX;
  unsigned thr = st[b].thr;
  float ls = 0.f, lc = 0.f;
  int stride = gridDim.x * blockDim.x;
  for (int idx = blockIdx.x * blockDim.x + threadIdx.x; idx < NPIX; idx += stride) {
    float v = gm_val(sb, idx >> 9, idx & 511);
    if (__float_as_uint(v) >= thr) { ls += v; lc += 1.f; }
  }
  float s = block_sum2(ls, 0.f, lds8);
  float c = block_sum2(lc, 0.f, lds8);
  if (threadIdx.x == 0) { atomicAdd(&es_sum[b], (double)s); atomicAdd(&es_cnt[b], (double)c); }
}

// ---------------- part 3: change point ----------------
__global__ void cp_collect(const float* __restrict__ seg, const SelState* __restrict__ st,
                           int* selCnt, int* tieCnt, int* sel, int* ties) {
  int b = blockIdx.y;
  const float* sb = seg + (size_t)b * NPIX;
  unsigned thr = st[b].thr;
  int stride = gridDim.x * blockDim.x;
  for (int idx = blockIdx.x * blockDim.x + threadIdx.x; idx < NPIX; idx += stride) {
    unsigned bits = __float_as_uint(edge_val(sb, idx >> 9, idx & 511));
    if (bits > thr) {
      int p = atomicAdd(&selCnt[b], 1);
      if (p < 64) sel[b * 64 + p] = idx;
    } else if (bits == thr) {
      int p = atomicAdd(&tieCnt[b], 1);
      if (p < TIE_CAP) ties[b * TIE_CAP + p] = idx;
    }
  }
}

// take the (k - G) smallest indices among ties => matches lax.top_k tie-breaking
__global__ void cp_ties_kernel(const SelState* __restrict__ st, const int* __restrict__ tieCnt,
                               const int* __restrict__ selCnt, int* sel,
                               const int* __restrict__ ties, int* selFinal) {
  __shared__ int arr[TIE_CAP];
  __shared__ int red[256];
  int b = blockIdx.x;
  int T = tieCnt[b]; if (T > TIE_CAP) T = TIE_CAP;
  int m = st[b].ties; if (m > CP_K) m = CP_K;
  int base = selCnt[b]; if (base > CP_K) base = CP_K;
  for (int i = threadIdx.x; i < T; i += blockDim.x) arr[i] = ties[b * TIE_CAP + i];
  __syncthreads();
  int rounds = (m < T) ? m : T;
  for (int r = 0; r < rounds; ++r) {
    int lm = 0x7fffffff;
    for (int i = threadIdx.x; i < T; i += blockDim.x) lm = min(lm, arr[i]);
    red[threadIdx.x] = lm;
    __syncthreads();
    for (int off = 128; off > 0; off >>= 1) {
      if ((int)threadIdx.x < off) red[threadIdx.x] = min(red[threadIdx.x], red[threadIdx.x + off]);
      __syncthreads();
    }
    int v = red[0];
    if (threadIdx.x == 0 && base + r < 64) sel[b * 64 + base + r] = v;
    for (int i = threadIdx.x; i < T; i += blockDim.x) if (arr[i] == v) arr[i] = 0x7fffffff;
    __syncthreads();
  }
  if (threadIdx.x == 0) {
    int tot = base + rounds;
    selFinal[b] = (tot > CP_K) ? CP_K : tot;
  }
}

// one wave per (sample, edge); lane == channel
__global__ void cp_score(const float* __restrict__ feat, const int* __restrict__ sel,
                         const int* __restrict__ selFinal, double* cp_sum, double* cp_cnt) {
  int b = blockIdx.y, t = blockIdx.x;
  if (t >= selFinal[b]) return;                       // block-uniform
  int idx = sel[b * 64 + t];
  int i = idx >> 9, j = idx & 511;
  if (!(i >= WIN && i < Hh - WIN && j >= WIN && j < Ww - WIN)) return;  // block-uniform
  int c = threadIdx.x;                                // 0..31 == channel
  const float* fp = feat + ((size_t)(b * Cc + c) * Hh) * Ww;
  float bsum = 0.f, asum = 0.f;
  #pragma unroll
  for (int dh = 0; dh < WIN; ++dh) {
    const float* rb = fp + (i - WIN + dh) * Ww + (j - WIN);
    const float* ra = fp + (i + dh) * Ww + j;
    #pragma unroll
    for (int dw = 0; dw < WIN; ++dw) { bsum += rb[dw]; asum += ra[dw]; }
  }
  float bm = bsum * (1.f / 25.f), am = asum * (1.f / 25.f);
  float bss = 0.f, ass = 0.f;
  #pragma unroll
  for (int dh = 0; dh < WIN; ++dh) {
    const float* rb = fp + (i - WIN + dh) * Ww + (j - WIN);
    const float* ra = fp + (i + dh) * Ww + j;
    #pragma unroll
    for (int dw = 0; dw < WIN; ++dw) {
      float db = rb[dw] - bm, da = ra[dw] - am;
      bss += db * db; ass += da * da;
    }
  }
  float bs  = sqrtf(bss * (1.f / 24.f)) + EPSF;
  float as_ = sqrtf(ass * (1.f / 24.f)) + EPSF;
  float sc = fabsf(am - bm) / (bs + as_);
  float tot = wave_sum2(sc, 0.f);                     // exact 32-lane sum via WMMA
  if (c == 0) {
    atomicAdd(&cp_sum[b], (double)(tot * (1.f / 32.f)));
    atomicAdd(&cp_cnt[b], 1.0);
  }
}

// ---------------- combine ----------------
__global__ void final_kernel(const double* __restrict__ acc, float* out) {
  const double* ic_sum = acc + 0;  const double* ic_sumsq = acc + 4;
  const double* es_sum = acc + 8;  const double* es_cnt   = acc + 12;
  const double* cp_sum = acc + 16; const double* cp_cnt   = acc + 20;
  double N = (double)NPIX;
  double ic = 0.0, es = 0.0, cp = 0.0;
  for (int b = 0; b < Bb; ++b) {
    double s = ic_sum[b], ss = ic_sumsq[b];
    ic += (ss - s * s / N) / (N - 1.0);
    es += es_sum[b] / es_cnt[b];
    double cnt = cp_cnt[b];
    cp += (cnt > 0.0) ? cp_sum[b] / fmax(cnt, 1.0) : 0.0;
  }
  ic *= 0.25; es *= 0.25; cp *= 0.25;
  out[0] = (float)(1.0 * ic - 2.0 * es + 1.5 * cp);
}

extern "C" void kernel_launch(void* const* d_in, const int* in_sizes, int n_in,
                              void* d_out, int out_size, void* d_ws, size_t ws_size,
                              hipStream_t stream) {
  (void)in_sizes; (void)n_in; (void)out_size; (void)ws_size;
  const float* seg  = (const float*)d_in[0];   // (4,1,512,512)
  const float* hic  = (const float*)d_in[1];   // (4,1,512,512)
  const float* feat = (const float*)d_in[2];   // (4,32,512,512)
  float* out = (float*)d_out;

  char* w = (char*)d_ws;
  double*   acc      = (double*)(w + 0);       // 24 doubles
  SelState* es_st    = (SelState*)(w + 192);
  SelState* cp_st    = (SelState*)(w + 256);
  int*      selCnt   = (int*)(w + 320);
  int*      tieCnt   = (int*)(w + 336);
  int*      selFinal = (int*)(w + 352);
  int*      sel      = (int*)(w + 384);        // 4*64 ints
  unsigned* hist     = (unsigned*)(w + 2048);  // 4*2048 u32
  int*      ties     = (int*)(w + 34816);      // 4*2048 ints

  double* ic_sum = acc + 0,  *ic_sumsq = acc + 4;
  double* es_sum = acc + 8,  *es_cnt   = acc + 12;
  double* cp_sum = acc + 16, *cp_cnt   = acc + 20;

  dim3 g64(64, 4);

  init_kernel<<<1, 256, 0, stream>>>(acc, es_st, cp_st, hist, selCnt, tieCnt, selFinal);

  // part 1
  ic_kernel<<<Bb * Hh, 256, 0, stream>>>(seg, hic, ic_sum, ic_sumsq);

  // part 2: exact top-k threshold on gm, then masked mean
  hist_kernel<0, 0><<<g64, 256, 0, stream>>>(seg, hist, es_st);
  scan_kernel<0><<<1, 128, 0, stream>>>(hist, es_st);
  hist_kernel<0, 1><<<g64, 256, 0, stream>>>(seg, hist, es_st);
  scan_kernel<1><<<1, 128, 0, stream>>>(hist, es_st);
  hist_kernel<0, 2><<<g64, 256, 0, stream>>>(seg, hist, es_st);
  scan_kernel<2><<<1, 128, 0, stream>>>(hist, es_st);
  es_final<<<g64, 256, 0, stream>>>(seg, es_st, es_sum, es_cnt);

  // part 3: exact top-51 indices on edge map, then window scores
  hist_kernel<1, 0><<<g64, 256, 0, stream>>>(seg, hist, cp_st);
  scan_kernel<0><<<1, 128, 0, stream>>>(hist, cp_st);
  hist_kernel<1, 1><<<g64, 256, 0, stream>>>(seg, hist, cp_st);
  scan_kernel<1><<<1, 128, 0, stream>>>(hist, cp_st);
  hist_kernel<1, 2><<<g64, 256, 0, stream>>>(seg, hist, cp_st);
  scan_kernel<2><<<1, 128, 0, stream>>>(hist, cp_st);
  cp_collect<<<g64, 256, 0, stream>>>(seg, cp_st, selCnt, tieCnt, sel, ties);
  cp_ties_kernel<<<Bb, 256, 0, stream>>>(cp_st, tieCnt, selCnt, sel, ties, selFinal);
  cp_score<<<dim3(CP_K, Bb), 32, 0, stream>>>(feat, sel, selFinal, cp_sum, cp_cnt);

  final_kernel<<<1, 1, 0, stream>>>(acc, out);
}